// RoutingConv_4071628997318
// MI455X (gfx1250) — compile-verified
//
#include <hip/hip_runtime.h>
#include <cstdint>

// Problem constants (match reference): N=50000 nodes, M=32 neighbors, D=128 feats,
// K=8 capsules, dd=16, residual mix a=0.9.
#define MNBR    32
#define DFEAT   128
#define KCAPS   8
#define ZSTRIDE 132          // floats per z row in LDS: 528B, 16B aligned, bank stride 4 (2-way worst)
#define WSTRIDE 9            // w row stride (conflict-free scatter writes, gcd(9,64)=1)
#define AMIX    0.9f

struct __align__(16) Smem {
    float z[2][MNBR][ZSTRIDE];   // per-wave gathered neighbor rows (16.9KB/node)
    float att_hi[DFEAT];         // att[D:2D], block-shared
    float w[2][MNBR][WSTRIDE];   // e_prime[j][k] per wave
};

__global__ __launch_bounds__(64) void routing_conv_kernel(
    const float* __restrict__ x,
    const float* __restrict__ att,
    const int*   __restrict__ neighbors,
    float*       __restrict__ out,
    int nNodes)
{
    __shared__ Smem sm;

    const int lane = threadIdx.x & 31;
    const int wv   = threadIdx.x >> 5;                  // wave in block: 0..1
    const int node = blockIdx.x * 2 + wv;
    const bool act = node < nNodes;

    // Stage att_hi (128 floats) cooperatively with 64 threads.
    sm.att_hi[threadIdx.x]      = att[DFEAT + threadIdx.x];
    sm.att_hi[threadIdx.x + 64] = att[DFEAT + threadIdx.x + 64];

    float4 xi = make_float4(0.f, 0.f, 0.f, 0.f);
    float  e_self = 0.f;

    if (act) {
        // Each lane owns neighbor index `lane`.
        const int nb = neighbors[(size_t)node * MNBR + lane];

        // Uniform 64-bit base for GVS-mode async loads (saddr = x).
        const unsigned long long xbase    = (unsigned long long)(uintptr_t)x;
        const uint32_t           lane_off = (uint32_t)(lane * 16);

        // ---- CDNA5 async gather: 32 rows x 512B straight into this wave's LDS slab.
        // GVS form: vdst = LDS byte offset, vaddr = 32-bit byte offset, saddr = base.
        // Fully unrolled: all 32 requests in flight (ASYNCcnt is 6 bits) before any wait.
        // j is compile-time constant -> readlane puts the row index in an SGPR
        // (scalar multiply + scalar guard; no ds_bpermute in the gather burst).
        #pragma unroll
        for (int j = 0; j < MNBR; ++j) {
            const int row = __builtin_amdgcn_readlane(nb, j);   // wave-uniform row
            // Low 32 bits of the generic pointer == wave-relative LDS byte offset
            // (LDS aperture is 4GB-aligned).
            const uint32_t lds_off =
                (uint32_t)(uintptr_t)&sm.z[wv][j][lane * 4];
            if (row < nNodes) {
                const uint32_t voff = (uint32_t)row * (DFEAT * 4u) + lane_off;
                asm volatile("global_load_async_to_lds_b128 %0, %1, %2"
                             :: "v"(lds_off), "v"(voff), "s"(xbase)
                             : "memory");
            } else {
                // reference pads with a zero row at index N
                *(float4*)&sm.z[wv][j][lane * 4] = make_float4(0.f, 0.f, 0.f, 0.f);
            }
        }

        // ---- Overlap with the async gather: e_self = x_i . att[:D]
        xi = *(const float4*)(x + (size_t)node * DFEAT + lane * 4);
        const float4 al = *(const float4*)(att + lane * 4);
        e_self = xi.x * al.x + xi.y * al.y + xi.z * al.z + xi.w * al.w;
        #pragma unroll
        for (int off = 16; off > 0; off >>= 1)
            e_self += __shfl_xor(e_self, off, 32);
    }

    __syncthreads();                                    // att_hi visible block-wide
    asm volatile("s_wait_asynccnt 0x0" ::: "memory");   // this wave's z slab resident

    if (act) {
        // ---- Pass 1: lane j walks neighbor row j out of LDS.
        //   e_j    = e_self + z_j . att_hi
        //   s[j,k] = sum over the 16 features of capsule k
        float e = e_self;
        float sk[KCAPS];
        #pragma unroll
        for (int k = 0; k < KCAPS; ++k) sk[k] = 0.f;

        #pragma unroll
        for (int c4 = 0; c4 < DFEAT / 4; ++c4) {
            const float4 z4 = *(const float4*)&sm.z[wv][lane][c4 * 4];
            const float4 a4 = *(const float4*)&sm.att_hi[c4 * 4];
            e += z4.x * a4.x + z4.y * a4.y + z4.z * a4.z + z4.w * a4.w;
            sk[c4 >> 2] += (z4.x + z4.y) + (z4.z + z4.w);   // c4*4/16 == c4/4
        }

        // ---- Softmax over the 32 neighbors (wave32 butterflies).
        float m = e;
        #pragma unroll
        for (int off = 16; off > 0; off >>= 1)
            m = fmaxf(m, __shfl_xor(m, off, 32));
        const float p = __expf(e - m);
        float ps = p;
        #pragma unroll
        for (int off = 16; off > 0; off >>= 1)
            ps += __shfl_xor(ps, off, 32);
        const float attn = p / ps;

        // e_prime[j][k] = attn_j * s[j][k]
        #pragma unroll
        for (int k = 0; k < KCAPS; ++k)
            sm.w[wv][lane][k] = attn * sk[k];
    }

    __syncthreads();                                    // w visible (cross-lane LDS dep)

    if (act) {
        // ---- Final: lane l owns features 4l..4l+3 (all inside capsule l/4).
        //   u[c] = a * sum_j e_prime[j][c/16] * z[j][c] + x_i[c]
        const int kk = lane >> 2;
        float4 acc = make_float4(0.f, 0.f, 0.f, 0.f);
        #pragma unroll 8
        for (int j = 0; j < MNBR; ++j) {
            const float  wt = sm.w[wv][j][kk];          // uniform row -> LDS broadcast
            const float4 z4 = *(const float4*)&sm.z[wv][j][lane * 4];
            acc.x += wt * z4.x;
            acc.y += wt * z4.y;
            acc.z += wt * z4.z;
            acc.w += wt * z4.w;
        }
        float4 o;
        o.x = AMIX * acc.x + xi.x;
        o.y = AMIX * acc.y + xi.y;
        o.z = AMIX * acc.z + xi.z;
        o.w = AMIX * acc.w + xi.w;
        *(float4*)(out + (size_t)node * DFEAT + lane * 4) = o;
    }
}

extern "C" void kernel_launch(void* const* d_in, const int* in_sizes, int n_in,
                              void* d_out, int out_size, void* d_ws, size_t ws_size,
                              hipStream_t stream) {
    (void)n_in; (void)d_ws; (void)ws_size; (void)out_size;
    const float* x         = (const float*)d_in[0];
    const float* att       = (const float*)d_in[1];
    const int*   neighbors = (const int*)d_in[2];
    // d_in[3] = max_iter: the routing variable p is dead in the reference -> unused.
    float* out = (float*)d_out;

    const int nNodes = in_sizes[0] / DFEAT;             // 50000
    const int blocks = (nNodes + 1) / 2;                // 2 nodes (2 waves) per block

    routing_conv_kernel<<<blocks, 64, 0, stream>>>(x, att, neighbors, out, nNodes);
}